// GNNDenoisingBlock_17712445129053
// MI455X (gfx1250) — compile-verified
//
#include <hip/hip_runtime.h>
#include <math.h>

#define HID 256
#define HEADS 4
#define HD 64
#define FFDIM 1024
#define LDSH_STRIDE 264   // 256 + 8 pad -> TDM pad_interval=7(256 dw), pad_amount=7(8 dw)
#define LDSF_STRIDE 1032  // 1024 + 8 pad

typedef __attribute__((ext_vector_type(2))) float v2f;
typedef __attribute__((ext_vector_type(8))) float v8f;
typedef __attribute__((ext_vector_type(4))) unsigned int v4u;
typedef __attribute__((ext_vector_type(4))) int v4i;
typedef __attribute__((ext_vector_type(8))) int v8i;

static __device__ __forceinline__ float warp_sum(float v) {
#pragma unroll
  for (int off = 16; off > 0; off >>= 1) v += __shfl_xor(v, off, 32);
  return v;
}

// float atomic-max via sign-split int/uint atomics (works with -inf init)
static __device__ __forceinline__ void atomic_max_f32(float* addr, float val) {
  if (val >= 0.0f) atomicMax((int*)addr, __float_as_int(val));
  else             atomicMin((unsigned int*)addr, __float_as_uint(val));
}

// ---------------------------------------------------------------------------
// QKV + skip projections: 4 GEMMs C = x[N,256] @ W[256,256] + b.
// One wave per 16x16 output tile, V_WMMA_F32_16X16X4_F32 along K.
// blockIdx.z selects which of {Wq,Wk,Wv,Wskip}.
// ---------------------------------------------------------------------------
__global__ void qkv_gemm_kernel(const float* __restrict__ x,
                                const float* __restrict__ Wq, const float* __restrict__ bq,
                                const float* __restrict__ Wk, const float* __restrict__ bk,
                                const float* __restrict__ Wv, const float* __restrict__ bv,
                                const float* __restrict__ Wsk, const float* __restrict__ bsk,
                                float* __restrict__ qo, float* __restrict__ ko,
                                float* __restrict__ vo, float* __restrict__ so,
                                int Nn)
{
  int which = blockIdx.z;
  const float* W    = (which == 0) ? Wq : (which == 1) ? Wk : (which == 2) ? Wv : Wsk;
  const float* bias = (which == 0) ? bq : (which == 1) ? bk : (which == 2) ? bv : bsk;
  float* out        = (which == 0) ? qo : (which == 1) ? ko : (which == 2) ? vo : so;

  const int tilesN = HID / 16;
  int mtiles = (Nn + 15) >> 4;
  int tile = blockIdx.x * 8 + (int)(threadIdx.x >> 5);
  if (tile >= mtiles * tilesN) return;           // wave-uniform exit, EXEC stays full
  int lane = threadIdx.x & 31;
  int tm = tile / tilesN, tn = tile % tilesN;
  int lg = lane >> 4;                            // lane-group -> K sub-index
  int li = lane & 15;
  int arow = tm * 16 + li;
  if (arow >= Nn) arow = Nn - 1;                 // clamp tail reads
  int ncol = tn * 16 + li;

  const float* ap = x + (size_t)arow * HID;
  v8f acc = {};
#pragma unroll 4
  for (int kk = 0; kk < HID; kk += 4) {
    int k0 = kk + lg * 2;
    v2f a, b;
    a.x = ap[k0];
    a.y = ap[k0 + 1];
    b.x = W[(size_t)k0 * HID + ncol];
    b.y = W[(size_t)(k0 + 1) * HID + ncol];
    acc = __builtin_amdgcn_wmma_f32_16x16x4_f32(false, a, false, b, (short)0, acc,
                                                false, false);
  }
  float bn = bias[ncol];
#pragma unroll
  for (int r = 0; r < 8; ++r) {
    int row = tm * 16 + lg * 8 + r;
    if (row < Nn) out[(size_t)row * HID + ncol] = acc[r] + bn;
  }
}

// ---------------------------------------------------------------------------
__global__ void seg_init_kernel(float* __restrict__ m, float* __restrict__ den, int count) {
  int i = blockIdx.x * blockDim.x + threadIdx.x;
  if (i < count) { m[i] = -INFINITY; den[i] = 0.0f; }
}

// per (edge, head): alpha = <q[dst,h], k[src,h]>/8 ; running segment max
__global__ void edge_alpha_kernel(const float* __restrict__ q, const float* __restrict__ k,
                                  const int* __restrict__ src, const int* __restrict__ dst,
                                  float* __restrict__ alpha, float* __restrict__ mmax, int E)
{
  int t = blockIdx.x * blockDim.x + threadIdx.x;
  if (t >= E * HEADS) return;
  int e = t >> 2, h = t & 3;
  int s = src[e], d = dst[e];
  const float4* qp = (const float4*)(q + (size_t)d * HID + h * HD);
  const float4* kp = (const float4*)(k + (size_t)s * HID + h * HD);
  float acc = 0.f;
#pragma unroll
  for (int i = 0; i < HD / 4; ++i) {
    float4 a = qp[i], b = kp[i];
    acc += a.x * b.x + a.y * b.y + a.z * b.z + a.w * b.w;
  }
  acc *= 0.125f;                                  // 1/sqrt(64)
  alpha[t] = acc;
  atomic_max_f32(&mmax[d * HEADS + h], acc);
}

// per (edge, head): ex = exp(alpha - max); segment-sum denominators; ex in place
__global__ void edge_exp_kernel(const int* __restrict__ dst,
                                float* __restrict__ alpha, const float* __restrict__ mmax,
                                float* __restrict__ den, int E)
{
  int t = blockIdx.x * blockDim.x + threadIdx.x;
  if (t >= E * HEADS) return;
  int e = t >> 2, h = t & 3;
  int d = dst[e];
  float ex = expf(alpha[t] - mmax[d * HEADS + h]);
  alpha[t] = ex;
  atomicAdd(&den[d * HEADS + h], ex);
}

// per (edge, feature): xnew[dst] += (ex/den) * v[src]   (xnew pre-loaded with skip GEMM)
__global__ void edge_scatter_kernel(const int* __restrict__ src, const int* __restrict__ dst,
                                    const float* __restrict__ ex, const float* __restrict__ den,
                                    const float* __restrict__ v, float* __restrict__ xnew,
                                    long long total)
{
  long long t = (long long)blockIdx.x * blockDim.x + threadIdx.x;
  if (t >= total) return;
  int e  = (int)(t >> 8);
  int hd = (int)(t & 255);
  int h  = hd >> 6;
  int s = src[e], d = dst[e];
  float w = ex[e * HEADS + h] / (den[d * HEADS + h] + 1e-16f);
  atomicAdd(&xnew[(size_t)d * HID + hd], w * v[(size_t)s * HID + hd]);
}

// h = layernorm(x + xnew) * g + b ; one wave per row
__global__ void ln1_kernel(const float* __restrict__ x, const float* __restrict__ xnew,
                           const float* __restrict__ g, const float* __restrict__ b,
                           float* __restrict__ hout, int Nn)
{
  int lane = threadIdx.x & 31;
  int row = blockIdx.x * 8 + (int)(threadIdx.x >> 5);
  if (row >= Nn) return;
  const float* xr = x + (size_t)row * HID;
  const float* nr = xnew + (size_t)row * HID;
  float vals[8];
  float s = 0.f;
#pragma unroll
  for (int i = 0; i < 8; ++i) {
    int c = lane + i * 32;
    vals[i] = xr[c] + nr[c];
    s += vals[i];
  }
  float mu = warp_sum(s) * (1.0f / HID);
  float vv = 0.f;
#pragma unroll
  for (int i = 0; i < 8; ++i) { float dd = vals[i] - mu; vv += dd * dd; }
  float inv = rsqrtf(warp_sum(vv) * (1.0f / HID) + 1e-5f);
#pragma unroll
  for (int i = 0; i < 8; ++i) {
    int c = lane + i * 32;
    hout[(size_t)row * HID + c] = (vals[i] - mu) * inv * g[c] + b[c];
  }
}

// ---------------------------------------------------------------------------
// Fused FFN: per 16-row tile, all in LDS:
//   h tile staged by the Tensor Data Mover (TENSOR_LOAD_TO_LDS) with HW padding
//   f = gelu(h @ W1 + b1)  (16x1024 in LDS)
//   y = h + f @ W2 + b2    (written back over h tile in LDS)
//   out = layernorm(y) * g2 + b2
// 8 waves per block; WMMA f32 16x16x4 for both GEMMs.
// ---------------------------------------------------------------------------
__global__ void ffn_fused_kernel(const float* __restrict__ h,
                                 const float* __restrict__ W1, const float* __restrict__ b1,
                                 const float* __restrict__ W2, const float* __restrict__ b2,
                                 const float* __restrict__ g2, const float* __restrict__ bb2,
                                 float* __restrict__ out, int Nn)
{
  extern __shared__ float smem[];
  float* lh = smem;                       // 16 x LDSH_STRIDE (264 = 256 + TDM pad)
  float* lf = smem + 16 * LDSH_STRIDE;    // 16 x LDSF_STRIDE

  int row0 = blockIdx.x * 16;
  int wave = threadIdx.x >> 5;
  int lane = threadIdx.x & 31;
  int lg = lane >> 4, li = lane & 15;

  // --- TDM stage: one wave issues a tensor DMA of the 16x256 h tile into LDS,
  // with hardware row padding (8 DWORDs after every 256 DWORDs -> stride 264).
  if (threadIdx.x < 32) {
    unsigned long long gaddr = (unsigned long long)(h + (size_t)row0 * HID);
    unsigned lds_base = __builtin_amdgcn_groupstaticsize();  // dynamic LDS offset

    v4u g0;
    g0.x = 1u;                                   // count=1, user mode, no gather
    g0.y = lds_base;                             // lds_addr[31:0]
    g0.z = (unsigned)(gaddr & 0xFFFFFFFFull);    // global_addr[31:0]
    g0.w = (unsigned)((gaddr >> 32) & 0x1FFFFFFull) | (2u << 30);  // addr[56:32] | type=2

    unsigned rows_left = (unsigned)(Nn - row0);  // OOB rows read as zero by TDM
    v8i g1;
    g1.s0 = (int)((2u << 16)        // data_size = 4 bytes
                | (1u << 20)        // pad_enable
                | (7u << 22)        // pad_interval: 256 DWORDs
                | (7u << 25));      // pad_amount: 8 DWORDs
    g1.s1 = (int)(((unsigned)HID & 0xFFFFu) << 16);          // tensor_dim0 lo16 (=256)
    g1.s2 = (int)((((unsigned)HID >> 16) & 0xFFFFu)          // tensor_dim0 hi16 (=0)
                | ((rows_left & 0xFFFFu) << 16));            // tensor_dim1 lo16
    g1.s3 = (int)(((rows_left >> 16) & 0xFFFFu)              // tensor_dim1 hi16
                | ((unsigned)HID << 16));                    // tile_dim0 = 256
    g1.s4 = (int)16u;                                        // tile_dim1 = 16, tile_dim2 = 0
    g1.s5 = (int)(unsigned)HID;                              // tensor_dim0_stride lo32 = 256
    g1.s6 = 0;                                               // stride hi, dim1_stride lo
    g1.s7 = 0;
    v4i gz4 = {0, 0, 0, 0};
    v8i gz8 = {0, 0, 0, 0, 0, 0, 0, 0};

    // 6-arg toolchain signature: (g0, g1, g2, g3, g4, cpol)
    __builtin_amdgcn_tensor_load_to_lds(g0, g1, gz4, gz4, gz8, 0);
    __builtin_amdgcn_s_wait_tensorcnt(0);        // issuing wave drains TENSORcnt
  }
  __syncthreads();

  // stage 1: 64 n-tiles over FF=1024, 8 per wave
  for (int nt = wave; nt < FFDIM / 16; nt += 8) {
    int ncol = nt * 16 + li;
    v8f acc = {};
#pragma unroll 4
    for (int kk = 0; kk < HID; kk += 4) {
      int k0 = kk + lg * 2;
      v2f a, b;
      a.x = lh[li * LDSH_STRIDE + k0];
      a.y = lh[li * LDSH_STRIDE + k0 + 1];
      b.x = W1[(size_t)k0 * FFDIM + ncol];
      b.y = W1[(size_t)(k0 + 1) * FFDIM + ncol];
      acc = __builtin_amdgcn_wmma_f32_16x16x4_f32(false, a, false, b, (short)0, acc,
                                                  false, false);
    }
    float bn = b1[ncol];
#pragma unroll
    for (int r = 0; r < 8; ++r) {
      float val = acc[r] + bn;
      val = 0.5f * val * (1.0f + erff(val * 0.70710678118654752f));  // exact gelu
      lf[(lg * 8 + r) * LDSF_STRIDE + ncol] = val;
    }
  }
  __syncthreads();

  // stage 2: 16 n-tiles over HID=256, 2 per wave; residual add, y -> lh in place
  for (int nt = wave; nt < HID / 16; nt += 8) {
    int ncol = nt * 16 + li;
    v8f acc = {};
#pragma unroll 4
    for (int kk = 0; kk < FFDIM; kk += 4) {
      int k0 = kk + lg * 2;
      v2f a, b;
      a.x = lf[li * LDSF_STRIDE + k0];
      a.y = lf[li * LDSF_STRIDE + k0 + 1];
      b.x = W2[(size_t)k0 * HID + ncol];
      b.y = W2[(size_t)(k0 + 1) * HID + ncol];
      acc = __builtin_amdgcn_wmma_f32_16x16x4_f32(false, a, false, b, (short)0, acc,
                                                  false, false);
    }
    float bn = b2[ncol];
#pragma unroll
    for (int r = 0; r < 8; ++r) {
      int rr = lg * 8 + r;
      float val = acc[r] + bn + lh[rr * LDSH_STRIDE + ncol];
      lh[rr * LDSH_STRIDE + ncol] = val;   // each (row,col) owned by exactly one lane
    }
  }
  __syncthreads();

  // LN2: wave handles rows {wave, wave+8}
  for (int r = wave; r < 16; r += 8) {
    int grow = row0 + r;
    if (grow >= Nn) continue;              // wave-uniform
    float vals[8];
    float s = 0.f;
#pragma unroll
    for (int i = 0; i < 8; ++i) { vals[i] = lh[r * LDSH_STRIDE + lane + i * 32]; s += vals[i]; }
    float mu = warp_sum(s) * (1.0f / HID);
    float vv = 0.f;
#pragma unroll
    for (int i = 0; i < 8; ++i) { float dd = vals[i] - mu; vv += dd * dd; }
    float inv = rsqrtf(warp_sum(vv) * (1.0f / HID) + 1e-5f);
#pragma unroll
    for (int i = 0; i < 8; ++i) {
      int c = lane + i * 32;
      out[(size_t)grow * HID + c] = (vals[i] - mu) * inv * g2[c] + bb2[c];
    }
  }
}

// ---------------------------------------------------------------------------
extern "C" void kernel_launch(void* const* d_in, const int* in_sizes, int n_in,
                              void* d_out, int out_size, void* d_ws, size_t ws_size,
                              hipStream_t stream)
{
  (void)n_in; (void)out_size; (void)ws_size;
  const float* x    = (const float*)d_in[0];
  const int*   edge = (const int*)d_in[1];
  const float* Wq = (const float*)d_in[2];  const float* bq = (const float*)d_in[3];
  const float* Wk = (const float*)d_in[4];  const float* bk = (const float*)d_in[5];
  const float* Wv = (const float*)d_in[6];  const float* bv = (const float*)d_in[7];
  const float* Wsk= (const float*)d_in[8];  const float* bsk= (const float*)d_in[9];
  const float* ln1g = (const float*)d_in[10]; const float* ln1b = (const float*)d_in[11];
  const float* W1 = (const float*)d_in[12]; const float* b1 = (const float*)d_in[13];
  const float* W2 = (const float*)d_in[14]; const float* b2 = (const float*)d_in[15];
  const float* ln2g = (const float*)d_in[16]; const float* ln2b = (const float*)d_in[17];
  float* out = (float*)d_out;

  int Nn = in_sizes[0] / HID;
  int E  = in_sizes[1] / 2;
  const int* src = edge;
  const int* dst = edge + E;

  float* ws = (float*)d_ws;
  size_t NH = (size_t)Nn * HID;
  float* q     = ws;                 // reused as h after attention is consumed
  float* kbuf  = ws + NH;
  float* vbuf  = ws + 2 * NH;
  float* xnew  = ws + 3 * NH;        // skip GEMM writes here; scatter accumulates
  float* alpha = ws + 4 * NH;        // E*H, ex overwrites in place
  float* mmax  = alpha + (size_t)E * HEADS;
  float* den   = mmax + (size_t)Nn * HEADS;
  float* hbuf  = q;

  int mtiles = (Nn + 15) / 16;
  int tiles = mtiles * (HID / 16);
  dim3 gq((tiles + 7) / 8, 1, 4);
  qkv_gemm_kernel<<<gq, 256, 0, stream>>>(x, Wq, bq, Wk, bk, Wv, bv, Wsk, bsk,
                                          q, kbuf, vbuf, xnew, Nn);

  int segc = Nn * HEADS;
  seg_init_kernel<<<(segc + 255) / 256, 256, 0, stream>>>(mmax, den, segc);

  int eh = E * HEADS;
  edge_alpha_kernel<<<(eh + 255) / 256, 256, 0, stream>>>(q, kbuf, src, dst, alpha, mmax, E);
  edge_exp_kernel<<<(eh + 255) / 256, 256, 0, stream>>>(dst, alpha, mmax, den, E);

  long long total = (long long)E * HID;
  edge_scatter_kernel<<<(unsigned)((total + 255) / 256), 256, 0, stream>>>(
      src, dst, alpha, den, vbuf, xnew, total);

  ln1_kernel<<<(Nn + 7) / 8, 256, 0, stream>>>(x, xnew, ln1g, ln1b, hbuf, Nn);

  size_t shbytes = (size_t)(16 * LDSH_STRIDE + 16 * LDSF_STRIDE) * sizeof(float);
  ffn_fused_kernel<<<mtiles, 256, shbytes, stream>>>(hbuf, W1, b1, W2, b2, ln2g, ln2b,
                                                     out, Nn);
}